// RBFLayer_4698694222589
// MI455X (gfx1250) — compile-verified
//
#include <hip/hip_runtime.h>
#include <math.h>

// ---------------------------------------------------------------------------
// Nerflet RBF layer for MI455X (gfx1250).
//   q[n,e] = F(p_n) . G_e  with F = [x2,y2,z2,xy,xz,yz,x,y,z,1,(0,0)] (K padded
//   to 12) computed as an fp32 WMMA GEMM using V_WMMA_F32_16X16X4_F32.
//   Everything else (exp / sigmoid / alpha blend) is VALU, streaming the
//   256 MB activation tensor once with coalesced non-temporal B128 loads.
// ---------------------------------------------------------------------------

typedef float v2f __attribute__((ext_vector_type(2)));
typedef float v4f __attribute__((ext_vector_type(4)));
typedef float v8f __attribute__((ext_vector_type(8)));

#define EC          64
#define NPTS        262144
#define KPAD        12          // 10 real features padded to 3 K=4 chunks
#define BLK         128         // 4 waves of 32
#define PTS_PER_BLK 128
#define QSTRIDE     65          // odd stride -> conflict-free column reads

// ---------------------------------------------------------------------------
// Setup: per-element quadratic-form coefficients G[e][12], |c_e|, and the
// bbox part of the penalty (initializes the penalty scalar).
// ---------------------------------------------------------------------------
__global__ __launch_bounds__(EC) void nerflet_setup(
    const float* __restrict__ constants, const float* __restrict__ centers,
    const float* __restrict__ radii,     const float* __restrict__ rotations,
    float* __restrict__ G, float* __restrict__ Cmag, float* __restrict__ pen_out)
{
    __shared__ float sred[EC];
    const int e = threadIdx.x;

    float dd[3];
    #pragma unroll
    for (int j = 0; j < 3; ++j) {
        float r = fabsf(radii[e * 3 + j]) + 0.005f;
        dd[j] = 1.0f / (r + 1e-8f);
    }
    const float rx = rotations[e * 3 + 0];
    const float ry = rotations[e * 3 + 1];
    const float rz = rotations[e * 3 + 2];
    const float cx = cosf(rx), sx = sinf(rx);
    const float cy = cosf(ry), sy = sinf(ry);
    const float cz = cosf(rz), sz = sinf(rz);
    const float R[3][3] = {
        { cz * cy, cz * sy * sx - sz * cx, cz * sy * cx + sz * sx },
        { sz * cy, sz * sy * sx + cz * cx, sz * sy * cx - cz * sx },
        { -sy,     cy * sx,                cy * cx                } };

    float A[3][3];
    #pragma unroll
    for (int i = 0; i < 3; ++i)
        #pragma unroll
        for (int k = 0; k < 3; ++k)
            A[i][k] = R[i][0] * dd[0] * R[k][0]
                    + R[i][1] * dd[1] * R[k][1]
                    + R[i][2] * dd[2] * R[k][2];

    const float mu[3] = { centers[e * 3], centers[e * 3 + 1], centers[e * 3 + 2] };
    float Amu[3];
    #pragma unroll
    for (int i = 0; i < 3; ++i)
        Amu[i] = A[i][0] * mu[0] + A[i][1] * mu[1] + A[i][2] * mu[2];
    const float h = mu[0] * Amu[0] + mu[1] * Amu[1] + mu[2] * Amu[2];

    float* g = G + e * KPAD;
    g[0] = A[0][0];       g[1] = A[1][1];       g[2] = A[2][2];
    g[3] = 2.f * A[0][1]; g[4] = 2.f * A[0][2]; g[5] = 2.f * A[1][2];
    g[6] = -2.f * Amu[0]; g[7] = -2.f * Amu[1]; g[8] = -2.f * Amu[2];
    g[9] = h;             g[10] = 0.f;          g[11] = 0.f;

    Cmag[e] = fabsf(constants[e]);

    const float bmin[3] = { -0.6f, -0.6f, -0.35f };
    const float bmax[3] = {  0.6f,  0.6f,  0.6f  };
    float p = 0.f;
    #pragma unroll
    for (int j = 0; j < 3; ++j)
        p += fmaxf(mu[j] - bmax[j], 0.f) + fmaxf(bmin[j] - mu[j], 0.f);
    sred[e] = p;
    __syncthreads();
    #pragma unroll
    for (int s = 32; s > 0; s >>= 1) {
        if (e < s) sred[e] += sred[e + s];
        __syncthreads();
    }
    if (e == 0) *pen_out = sred[0];   // init (bbox term); main kernel atomicAdds
}

// ---------------------------------------------------------------------------
// Main kernel.
// ---------------------------------------------------------------------------
__global__ __launch_bounds__(BLK) void nerflet_main(
    const float* __restrict__ wsp,  const float* __restrict__ act,
    const float* __restrict__ dists,
    const float* __restrict__ G,    const float* __restrict__ Cmag,
    float* __restrict__ out, float* __restrict__ pen_out)
{
    __shared__ float qlds[PTS_PER_BLK * QSTRIDE];   // ~32.5 KB
    __shared__ float pred[BLK];

    const int  lane = threadIdx.x & 31;
    const int  wave = threadIdx.x >> 5;
    const bool hi   = (lane & 16) != 0;   // ISA A/B layout: lanes 16-31 = K+2
    const int  l16  = lane & 15;
    const int  blockBase = blockIdx.x * PTS_PER_BLK;

    // ---- phase 1: q[point, elt] via fp32 WMMA (16x16x4, K padded to 12) ----
    // Each wave owns 32 points = two 16-row M-tiles; 4 N-tiles cover EC=64.
    // A operand per K-chunk (no dynamic indexing -> just 2 cndmask per v2f):
    //   kc=0: K{0,1}={x2,y2}   K{2,3}={z2,xy}
    //   kc=1: K{4,5}={xz,yz}   K{6,7}={x, y }
    //   kc=2: K{8,9}={z, 1 }   K{10,11}={0,0}
    v2f aop[2][3];
    #pragma unroll
    for (int mt = 0; mt < 2; ++mt) {
        const int p = blockBase + wave * 32 + mt * 16 + l16;
        const float x = wsp[3 * p + 0];
        const float y = wsp[3 * p + 1];
        const float z = wsp[3 * p + 2];
        aop[mt][0].x = hi ? z * z : x * x;
        aop[mt][0].y = hi ? x * y : y * y;
        aop[mt][1].x = hi ? x     : x * z;
        aop[mt][1].y = hi ? y     : y * z;
        aop[mt][2].x = hi ? 0.f   : z;
        aop[mt][2].y = hi ? 0.f   : 1.f;
    }

    // B operand: lane supplies G[e][k] for e = nt*16 + l16, K pair by half.
    const int koff = hi ? 2 : 0;
    v2f bop[4][3];
    #pragma unroll
    for (int nt = 0; nt < 4; ++nt) {
        const int e = nt * 16 + l16;
        #pragma unroll
        for (int kc = 0; kc < 3; ++kc)
            bop[nt][kc] = *(const v2f*)(G + e * KPAD + kc * 4 + koff);
    }

    #pragma unroll
    for (int mt = 0; mt < 2; ++mt) {
        #pragma unroll
        for (int nt = 0; nt < 4; ++nt) {
            v8f acc = {0.f, 0.f, 0.f, 0.f, 0.f, 0.f, 0.f, 0.f};
            #pragma unroll
            for (int kc = 0; kc < 3; ++kc)
                acc = __builtin_amdgcn_wmma_f32_16x16x4_f32(
                        false, aop[mt][kc], false, bop[nt][kc],
                        (short)0, acc, false, false);
            // C layout: VGPR r -> M = r (+8 for lanes>=16), N = l16
            const int mrow = wave * 32 + mt * 16 + (hi ? 8 : 0);
            #pragma unroll
            for (int r = 0; r < 8; ++r)
                qlds[(mrow + r) * QSTRIDE + nt * 16 + l16] = acc[r];
        }
    }
    __syncthreads();

    // ---- phase 2: one thread per point ----
    const int t = threadIdx.x;
    const int n = blockBase + t;
    float* qrow = &qlds[t * QSTRIDE];   // stride 65 -> conflict-free

    float sum = 0.f, pen = 0.f;
    #pragma unroll 8
    for (int e = 0; e < EC; ++e) {
        const float rbf = __expf(-0.5f * qrow[e]) * Cmag[e];
        pen += fmaxf(rbf - 0.01f, 0.f);
        sum += rbf;
        qrow[e] = rbf;                  // reuse slot for weights pass
    }
    const float inv  = 1.0f / (sum + 1e-6f);
    const float dist = dists[n];

    float o0 = 0.f, o1 = 0.f, o2 = 0.f, o3 = 0.f;
    #pragma unroll 4
    for (int e = 0; e < EC; ++e) {
        const float w = qrow[e] * inv;  // rbf >= 0 always, KILL_THRESH=0 is a no-op
        const v4f a = __builtin_nontemporal_load(
            (const v4f*)(act + (((size_t)e * NPTS + (size_t)n) << 2)));
        const float r0 = 1.0f / (1.0f + __expf(-a.x));
        const float r1 = 1.0f / (1.0f + __expf(-a.y));
        const float r2 = 1.0f / (1.0f + __expf(-a.z));
        const float al = 1.0f - __expf(-fmaxf(a.w, 0.f) * dist);
        o0 += w * r0; o1 += w * r1; o2 += w * r2; o3 += w * al;
    }
    v4f o = { o0, o1, o2, o3 };
    __builtin_nontemporal_store(o, (v4f*)(out + ((size_t)n << 2)));

    // penalty: block reduce then one atomic
    pred[t] = pen;
    __syncthreads();
    #pragma unroll
    for (int s = BLK / 2; s > 0; s >>= 1) {
        if (t < s) pred[t] += pred[t + s];
        __syncthreads();
    }
    if (t == 0) atomicAdd(pen_out, pred[0] * (0.001f / (float)NPTS));
}

// ---------------------------------------------------------------------------
extern "C" void kernel_launch(void* const* d_in, const int* in_sizes, int n_in,
                              void* d_out, int out_size, void* d_ws, size_t ws_size,
                              hipStream_t stream)
{
    const float* wsp       = (const float*)d_in[0];
    const float* act       = (const float*)d_in[1];
    const float* dists     = (const float*)d_in[2];
    const float* constants = (const float*)d_in[3];
    const float* centers   = (const float*)d_in[4];
    const float* radii     = (const float*)d_in[5];
    const float* rotations = (const float*)d_in[6];

    float* out  = (float*)d_out;
    float* pen  = out + (size_t)NPTS * 4;    // penalty scalar after [N,4]
    float* G    = (float*)d_ws;              // EC*KPAD floats
    float* Cmag = G + EC * KPAD;             // EC floats

    nerflet_setup<<<1, EC, 0, stream>>>(constants, centers, radii, rotations,
                                        G, Cmag, pen);
    nerflet_main<<<NPTS / PTS_PER_BLK, BLK, 0, stream>>>(
        wsp, act, dists, G, Cmag, out, pen);
}